// DiffGCL_Encoder_16724602651076
// MI455X (gfx1250) — compile-verified
//
#include <hip/hip_runtime.h>
#include <hip/hip_bf16.h>

#define N_USERS 50000
#define N_ITEMS 100000
#define N_TOTAL (N_USERS + N_ITEMS)   // 150000
#define EMB 64
#define NNZ_CNT 2000000
#define RANK 6

typedef float v2f  __attribute__((ext_vector_type(2)));
typedef float v8f  __attribute__((ext_vector_type(8)));

// ---------------------------------------------------------------------------
// 1) init: ACC = ego, TMP1 = ego (spmm input), CUR = 0, core scratch = 0
//    ego = concat(user_ori, item_ori); all float4 (b128) vectorized.
// ---------------------------------------------------------------------------
__global__ __launch_bounds__(256) void init_kernel(
    const float4* __restrict__ user_ori4, const float4* __restrict__ item_ori4,
    float4* __restrict__ acc4, float4* __restrict__ tmp4,
    float4* __restrict__ cur4, float4* __restrict__ core_ws4) {
  const int n4 = N_TOTAL * EMB / 4;            // 2,400,000
  int i = blockIdx.x * blockDim.x + threadIdx.x;
  if (i < n4) {
    const int u4 = N_USERS * EMB / 4;          // 800,000
    float4 e = (i < u4) ? user_ori4[i] : item_ori4[i - u4];
    acc4[i] = e;
    tmp4[i] = e;
    cur4[i] = make_float4(0.f, 0.f, 0.f, 0.f);
  }
  if (i < 448) core_ws4[i] = make_float4(0.f, 0.f, 0.f, 0.f); // 1792 floats of core scratch
}

// ---------------------------------------------------------------------------
// 2) SpMM: out[row] += val * in[col], one wave per edge (grid-strided).
//    Each wave: 2 coalesced 128B row loads + 64 global_atomic_add_f32
//    (non-returning, so tracked on STOREcnt only).
// ---------------------------------------------------------------------------
__global__ __launch_bounds__(256) void spmm_kernel(
    const int* __restrict__ rows, const int* __restrict__ cols,
    const float* __restrict__ vals, const float* __restrict__ xin,
    float* __restrict__ xout) {
  const int lane   = threadIdx.x & 31;
  const int wid    = (blockIdx.x * blockDim.x + threadIdx.x) >> 5;
  const int nwaves = (gridDim.x * blockDim.x) >> 5;
  for (int e = wid; e < NNZ_CNT; e += nwaves) {
    const int   r = rows[e];
    const int   c = cols[e];
    const float v = vals[e];
    const float* src = xin + c * EMB;
    float x0 = src[lane];
    float x1 = src[lane + 32];
    float* dst = xout + r * EMB;
    atomicAdd(dst + lane,      v * x0);
    atomicAdd(dst + lane + 32, v * x1);
  }
}

// ---------------------------------------------------------------------------
// 3) acc += cur; zero next spmm target (b128 streams)
// ---------------------------------------------------------------------------
__global__ __launch_bounds__(256) void acc_add_zero_kernel(
    float4* __restrict__ acc4, const float4* __restrict__ cur4,
    float4* __restrict__ next_zero4) {
  const int n4 = N_TOTAL * EMB / 4;
  int i = blockIdx.x * blockDim.x + threadIdx.x;
  if (i >= n4) return;
  float4 a = acc4[i], c = cur4[i];
  a.x += c.x; a.y += c.y; a.z += c.z; a.w += c.w;
  acc4[i] = a;
  next_zero4[i] = make_float4(0.f, 0.f, 0.f, 0.f);
}

// ---------------------------------------------------------------------------
// 4) mean: acc = (acc + last) / (N_LAYERS+1)
// ---------------------------------------------------------------------------
__global__ __launch_bounds__(256) void finalize_mean_kernel(
    float4* __restrict__ acc4, const float4* __restrict__ last4) {
  const int n4 = N_TOTAL * EMB / 4;
  int i = blockIdx.x * blockDim.x + threadIdx.x;
  if (i >= n4) return;
  const float inv = 1.0f / 3.0f;
  float4 a = acc4[i], c = last4[i];
  a.x = (a.x + c.x) * inv; a.y = (a.y + c.y) * inv;
  a.z = (a.z + c.z) * inv; a.w = (a.w + c.w) * inv;
  acc4[i] = a;
}

// ---------------------------------------------------------------------------
// 5) core_pre[r][d] = sum_i v[i][r] * feat[i][d]   (RANK x EMB = 6x64)
//    Block: 4 row-streams x 64 dims; LDS reduce; 384 atomics per block.
// ---------------------------------------------------------------------------
__global__ __launch_bounds__(256) void corepre_kernel(
    const float* __restrict__ feat, const float* __restrict__ vm,
    float* __restrict__ core_pre, int Mr) {
  __shared__ float red[4][RANK][EMB];
  const int g = threadIdx.x >> 6;     // 0..3 row-stream within block
  const int d = threadIdx.x & 63;     // emb dim
  float acc[RANK];
#pragma unroll
  for (int r = 0; r < RANK; ++r) acc[r] = 0.f;
  for (int row = blockIdx.x * 4 + g; row < Mr; row += gridDim.x * 4) {
    float f = feat[row * EMB + d];
#pragma unroll
    for (int r = 0; r < RANK; ++r) acc[r] += vm[row * RANK + r] * f;
  }
#pragma unroll
  for (int r = 0; r < RANK; ++r) red[g][r][d] = acc[r];
  __syncthreads();
  if (threadIdx.x < 64) {
#pragma unroll
    for (int r = 0; r < RANK; ++r) {
      float s = red[0][r][threadIdx.x] + red[1][r][threadIdx.x] +
                red[2][r][threadIdx.x] + red[3][r][threadIdx.x];
      atomicAdd(&core_pre[r * EMB + threadIdx.x], s);
    }
  }
}

// ---------------------------------------------------------------------------
// 6) core[r][j] = s[r] * sum_k core_pre[r][k] * W[j][k]; pad rows 6,7 -> 0.
//    Stored as 8x64 so the WMMA B-fragment needs no bounds checks.
// ---------------------------------------------------------------------------
__global__ __launch_bounds__(512) void corescale_kernel(
    const float* __restrict__ core_pre, const float* __restrict__ W,
    const float* __restrict__ s, float* __restrict__ core8) {
  const int r = threadIdx.x >> 6;   // 0..7
  const int j = threadIdx.x & 63;
  float val = 0.f;
  if (r < RANK) {
    float a = 0.f;
#pragma unroll
    for (int k = 0; k < EMB; ++k) a += core_pre[r * EMB + k] * W[j * EMB + k];
    val = s[r] * a;
  }
  core8[r * EMB + j] = val;
}

// ---------------------------------------------------------------------------
// 7) out = sigmoid(u @ core8)   (M x 6pad8 @ 8 x 64) via WMMA.
//    One wave per 16x16 output tile; K=8 in two v_wmma_f32_16x16x4_f32 steps.
//    All u-loads are unconditionally in-bounds (k in 0..5); the K=6,7 zero
//    padding is applied with v_cndmask on loaded values, so no exec-mask
//    branches around the fragment loads. Tail exit is wave-uniform -> EXEC
//    is all-ones at every WMMA.
// ---------------------------------------------------------------------------
__global__ __launch_bounds__(256) void diffout_wmma_kernel(
    const float* __restrict__ u, const float* __restrict__ core8,
    float* __restrict__ out, int M) {
  const int wave  = (blockIdx.x * blockDim.x + threadIdx.x) >> 5;
  const int tiles = (M >> 4) * (EMB / 16);     // M/16 row-tiles x 4 col-tiles
  if (wave >= tiles) return;                   // uniform across the wave
  const int lane = threadIdx.x & 31;
  const int half = lane >> 4;                  // 0: lanes 0-15, 1: lanes 16-31
  const int l16  = lane & 15;
  const int m0 = (wave >> 2) * 16;
  const int n0 = (wave & 3) * 16;
  const int mrow = m0 + l16;                   // A-matrix row held by this lane
  const int n    = n0 + l16;                   // B/C/D column held by this lane

  // ---- A fragments (16x4 per step): lanes 0-15 hold K={kb,kb+1},
  //      lanes 16-31 hold K={kb+2,kb+3}. kb=0 -> K in {0..3}: always valid.
  //      kb=4 -> lanes 0-15 K={4,5} valid, lanes 16-31 K={6,7} = zero pad.
  const float* urow = u + mrow * RANK;
  v2f a0, a1;
  a0.x = urow[half * 2];
  a0.y = urow[half * 2 + 1];
  float a4 = urow[4];
  float a5 = urow[5];
  a1.x = half ? 0.f : a4;                      // v_cndmask, no branch
  a1.y = half ? 0.f : a5;

  // ---- B fragments (4x16 per step): core8 has rows 6,7 pre-zeroed.
  v2f b0, b1;
  b0.x = core8[(half * 2    ) * EMB + n];
  b0.y = core8[(half * 2 + 1) * EMB + n];
  b1.x = core8[(4 + half * 2    ) * EMB + n];
  b1.y = core8[(4 + half * 2 + 1) * EMB + n];

  v8f c = {};
  c = __builtin_amdgcn_wmma_f32_16x16x4_f32(false, a0, false, b0, (short)0, c, false, false);
  c = __builtin_amdgcn_wmma_f32_16x16x4_f32(false, a1, false, b1, (short)0, c, false, false);

  // D layout: VGPR i -> row m0+i (lanes 0-15) / m0+8+i (lanes 16-31), col n
#pragma unroll
  for (int i = 0; i < 8; ++i) {
    const int row = m0 + (half ? 8 : 0) + i;
    // sigmoid: single v_exp_f32 + v_rcp_f32 per element
    out[row * EMB + n] = __builtin_amdgcn_rcpf(1.0f + __expf(-c[i]));
  }
}

// ---------------------------------------------------------------------------
// Launch
// ---------------------------------------------------------------------------
extern "C" void kernel_launch(void* const* d_in, const int* in_sizes, int n_in,
                              void* d_out, int out_size, void* d_ws, size_t ws_size,
                              hipStream_t stream) {
  const int*   rows     = (const int*)  d_in[0];
  const int*   cols     = (const int*)  d_in[1];
  const float* vals     = (const float*)d_in[2];
  const float* user_ori = (const float*)d_in[3];
  const float* item_ori = (const float*)d_in[4];
  const float* user_aug = (const float*)d_in[5];
  const float* item_aug = (const float*)d_in[6];
  const float* W        = (const float*)d_in[7];
  const float* u1       = (const float*)d_in[8];
  const float* s1       = (const float*)d_in[9];
  const float* v1       = (const float*)d_in[10];
  const float* u2       = (const float*)d_in[11];
  const float* s2       = (const float*)d_in[12];
  const float* v2       = (const float*)d_in[13];

  float* out = (float*)d_out;
  float* ACC  = out;                              // mean_emb (N x 64), final user/item_all_ori
  float* TMP1 = out + (size_t)N_TOTAL * EMB;      // aug region doubles as spmm ping buffer
  float* OUT_UA = out + (size_t)N_TOTAL * EMB;                         // user_all_aug
  float* OUT_IA = out + (size_t)N_TOTAL * EMB + (size_t)N_USERS * EMB; // item_all_aug

  float* WS    = (float*)d_ws;
  float* CUR   = WS;                              // N x 64 spmm pong buffer
  float* CP1   = WS + (size_t)N_TOTAL * EMB;      // 6 x 64
  float* CP2   = CP1 + RANK * EMB;                // 6 x 64
  float* CORE1 = CP2 + RANK * EMB;                // 8 x 64 (padded)
  float* CORE2 = CORE1 + 8 * EMB;                 // 8 x 64 (padded)

  const int n4      = N_TOTAL * EMB / 4;          // 2,400,000 float4
  const int blk_ew  = (n4 + 255) / 256;

  // LightGCN propagation: acc = ego + A*ego + A*A*ego, then /3
  init_kernel<<<blk_ew, 256, 0, stream>>>(
      (const float4*)user_ori, (const float4*)item_ori,
      (float4*)ACC, (float4*)TMP1, (float4*)CUR, (float4*)CP1);
  spmm_kernel<<<8192, 256, 0, stream>>>(rows, cols, vals, TMP1, CUR);
  acc_add_zero_kernel<<<blk_ew, 256, 0, stream>>>(
      (float4*)ACC, (const float4*)CUR, (float4*)TMP1);
  spmm_kernel<<<8192, 256, 0, stream>>>(rows, cols, vals, CUR, TMP1);
  finalize_mean_kernel<<<blk_ew, 256, 0, stream>>>(
      (float4*)ACC, (const float4*)TMP1);

  // Diffusion layers: core_pre = v^T @ feat; core = s * (core_pre @ W^T)
  corepre_kernel<<<256, 256, 0, stream>>>(item_aug, v1, CP1, N_ITEMS);
  corepre_kernel<<<256, 256, 0, stream>>>(user_aug, v2, CP2, N_USERS);
  corescale_kernel<<<1, 512, 0, stream>>>(CP1, W, s1, CORE1);
  corescale_kernel<<<1, 512, 0, stream>>>(CP2, W, s2, CORE2);

  // out = sigmoid(u @ core) via WMMA (overwrites TMP1 region last)
  {
    int tiles1 = (N_USERS / 16) * (EMB / 16);     // 12500
    int tiles2 = (N_ITEMS / 16) * (EMB / 16);     // 25000
    diffout_wmma_kernel<<<(tiles1 + 7) / 8, 256, 0, stream>>>(u1, CORE1, OUT_UA, N_USERS);
    diffout_wmma_kernel<<<(tiles2 + 7) / 8, 256, 0, stream>>>(u2, CORE2, OUT_IA, N_ITEMS);
  }
}